// AxisSimplestSpline_69724499083957
// MI455X (gfx1250) — compile-verified
//
#include <hip/hip_runtime.h>

// ---------------------------------------------------------------------------
// AxisSimplestSpline on gfx1250 (MI455X)
//   raw: [B,3,H,W] f32   ys: [B,8,16] f32   A: [B,3,8] f32  ->  out: [B,3,H,W]
//
// Memory-bound (24 B/pixel -> ~8.6us HBM floor). One streaming pass; the
// 3->8 projection runs on V_WMMA_F32_16X16X4_F32 (exact f32, K=4 fits the
// 3-channel contraction), spline + 8->3 backprojection run on VALU with all
// per-batch constants in registers/LDS.
//
// ws layout per batch (stride 256 floats):
//   [0..7]   mins
//   [8..15]  dx
//   [16..23] 1/dx
//   [24..47] A           (c*8 + a)
//   [48..71] pinv(A)     (a*3 + c)
//   [72..215] ys_full    (a*18 + i),  18 = mins | ys[16] | maxs
// ---------------------------------------------------------------------------

#define NAX   8
#define NKNOT 16
#define WSB   256        // floats per batch in workspace
#define EPSV  1e-4f

typedef float v2f __attribute__((ext_vector_type(2)));
typedef float v8f __attribute__((ext_vector_type(8)));

__global__ void spline_setup_kernel(const float* __restrict__ ys,
                                    const float* __restrict__ A,
                                    float* __restrict__ ws) {
  const int b    = blockIdx.x;
  const int lane = threadIdx.x;
  const float* Ab = A + b * 3 * NAX;
  float* w = ws + b * WSB;

  // every lane keeps the full 3x8 A in registers
  float Am[3][NAX];
#pragma unroll
  for (int c = 0; c < 3; ++c)
#pragma unroll
    for (int a = 0; a < NAX; ++a)
      Am[c][a] = Ab[c * NAX + a];

  // Gram matrix G = A A^T (3x3) and its inverse (adjugate / det)
  float G[3][3];
#pragma unroll
  for (int i = 0; i < 3; ++i)
#pragma unroll
    for (int j = 0; j < 3; ++j) {
      float s = 0.f;
#pragma unroll
      for (int a = 0; a < NAX; ++a) s += Am[i][a] * Am[j][a];
      G[i][j] = s;
    }
  const float det =
      G[0][0] * (G[1][1] * G[2][2] - G[1][2] * G[2][1]) -
      G[0][1] * (G[1][0] * G[2][2] - G[1][2] * G[2][0]) +
      G[0][2] * (G[1][0] * G[2][1] - G[1][1] * G[2][0]);
  const float id = 1.0f / det;
  float Gi[3][3];
  Gi[0][0] = (G[1][1] * G[2][2] - G[1][2] * G[1][2]) * id;
  Gi[0][1] = (G[0][2] * G[1][2] - G[0][1] * G[2][2]) * id;
  Gi[0][2] = (G[0][1] * G[1][2] - G[0][2] * G[1][1]) * id;
  Gi[1][1] = (G[0][0] * G[2][2] - G[0][2] * G[0][2]) * id;
  Gi[1][2] = (G[0][2] * G[0][1] - G[0][0] * G[1][2]) * id;
  Gi[2][2] = (G[0][0] * G[1][1] - G[0][1] * G[0][1]) * id;
  Gi[1][0] = Gi[0][1]; Gi[2][0] = Gi[0][2]; Gi[2][1] = Gi[1][2];

  if (lane < NAX) {
    const int a = lane;
    float mn = 0.f, mx = 0.f;
#pragma unroll
    for (int c = 0; c < 3; ++c) {
      mn += fminf(Am[c][a], 0.f);
      mx += fmaxf(Am[c][a], 0.f);
    }
    const float span = mx + EPSV - mn;
    const float dxv  = span / (float)(NKNOT + 1);   // 17 segments
    w[a]          = mn;
    w[8 + a]      = dxv;
    w[16 + a]     = 1.0f / dxv;
    // ys_full = [mins, ys..., maxs]
    w[72 + a * 18 + 0] = mn;
#pragma unroll
    for (int i = 0; i < NKNOT; ++i)
      w[72 + a * 18 + 1 + i] = ys[(b * NAX + a) * NKNOT + i];
    w[72 + a * 18 + 17] = mx;
    // pinv(A)[a][c] = sum_j A[j][a] * Gi[j][c]
#pragma unroll
    for (int c = 0; c < 3; ++c) {
      float p = 0.f;
#pragma unroll
      for (int j = 0; j < 3; ++j) p += Am[j][a] * Gi[j][c];
      w[48 + a * 3 + c] = p;
    }
  }
  if (lane < 24) w[24 + lane] = Ab[lane];   // copy A verbatim
}

// ---------------------------------------------------------------------------
// Main pass: block = 256 threads = 8 waves, each wave owns 32 pixels.
// Projection: two V_WMMA_F32_16X16X4_F32 tiles (axes on M via A-operand,
// pixels on N via B-operand), merged to a 32-lane pixel layout via
// ds_bpermute so spline + backprojection run on all lanes.
// ---------------------------------------------------------------------------
__global__ void __launch_bounds__(256)
spline_main_kernel(const float* __restrict__ raw,
                   const float* __restrict__ ws,
                   float* __restrict__ out,
                   int HWpix) {
  __shared__ float ysf[NAX * 18];

  const int b   = blockIdx.y;
  const int tid = threadIdx.x;
  const float* w = ws + b * WSB;

  if (tid < NAX * 18) ysf[tid] = w[72 + tid];
  __syncthreads();

  const int lane = tid & 31;
  const int wave = tid >> 5;
  const int base = blockIdx.x * 256 + wave * 32;    // first pixel of this wave
  if (base >= HWpix) return;                        // wave-uniform exit

  // per-lane copies of per-batch constants (uniform addresses -> s_loads)
  float mins[NAX], dxv[NAX], invdx[NAX], pv[NAX][3];
#pragma unroll
  for (int r = 0; r < NAX; ++r) {
    mins[r]  = w[r];
    dxv[r]   = w[8 + r];
    invdx[r] = w[16 + r];
    pv[r][0] = w[48 + r * 3 + 0];
    pv[r][1] = w[48 + r * 3 + 1];
    pv[r][2] = w[48 + r * 3 + 2];
  }

  // A-operand (16x4 f32): M = axis (rows 8..15 zero), K = channel (K=3 pad 0)
  //   vgpr0: lanes0-15 -> A[ch0][axis], lanes16-31 -> A[ch2][axis]
  //   vgpr1: lanes0-15 -> A[ch1][axis], lanes16-31 -> 0
  const int axm  = lane & 15;
  const int half = lane >> 4;
  v2f amat;
  amat.x = (axm < NAX) ? (half ? w[24 + 2 * 8 + axm] : w[24 + 0 * 8 + axm]) : 0.f;
  amat.y = (axm < NAX && !half) ? w[24 + 1 * 8 + axm] : 0.f;

  const float* rb = raw + (size_t)b * 3 * HWpix;
  const int pl = base + (lane & 15);

  // B-operand (4x16 f32): K = channel rows, N = pixel columns
  //   vgpr0: lanes0-15 -> ch0[pix], lanes16-31 -> ch2[pix]
  //   vgpr1: lanes0-15 -> ch1[pix], lanes16-31 -> 0 (K=3 pad)
  v8f cz = {};
  v2f b0, b1;
  b0.x = rb[(half ? 2 * (size_t)HWpix : 0) + pl];
  b0.y = half ? 0.f : rb[(size_t)HWpix + pl];
  b1.x = rb[(half ? 2 * (size_t)HWpix : 0) + pl + 16];
  b1.y = half ? 0.f : rb[(size_t)HWpix + pl + 16];

  // D[axis, pixel] = sum_ch A[ch][axis] * raw[ch][pixel]
  v8f d0 = __builtin_amdgcn_wmma_f32_16x16x4_f32(
      false, amat, false, b0, (short)0, cz, false, false);
  v8f d1 = __builtin_amdgcn_wmma_f32_16x16x4_f32(
      false, amat, false, b1, (short)0, cz, false, false);

  // Merge tiles: fx[r][lane] = f(axis r, pixel base+lane) for all 32 lanes.
  // d0 holds pixels base..base+15 in lanes 0-15; d1 holds base+16..base+31
  // in lanes 0-15 -> pull d1[lane&15] into the upper half via ds_bpermute.
  const int bpi = (lane & 15) << 2;
  float fx[NAX];
#pragma unroll
  for (int r = 0; r < NAX; ++r) {
    float hiv = __int_as_float(
        __builtin_amdgcn_ds_bpermute(bpi, __float_as_int(d1[r])));
    fx[r] = (lane < 16) ? d0[r] : hiv;
  }

  // Spline eval + back-projection (pinv is per-batch constant -> plain FMAs;
  // WMMA here would waste 13/16 columns since N=3).
  float oc0 = 0.f, oc1 = 0.f, oc2 = 0.f;
#pragma unroll
  for (int r = 0; r < NAX; ++r) {
    const float fv = fx[r];
    const float t  = (fv - mins[r]) * invdx[r];
    int idx = (int)t;
    idx = (idx < 0) ? 0 : ((idx > NKNOT) ? NKNOT : idx);
    const float ylo  = ysf[r * 18 + idx];
    const float yhi  = ysf[r * 18 + idx + 1];
    const float xshi = mins[r] + (float)(idx + 1) * dxv[r];
    const float est  = yhi - (xshi - fv) * (yhi - ylo) * invdx[r];
    oc0 += est * pv[r][0];
    oc1 += est * pv[r][1];
    oc2 += est * pv[r][2];
  }

  float* ob = out + (size_t)b * 3 * HWpix;
  const int p = base + lane;
  ob[p]                    = oc0;
  ob[(size_t)HWpix + p]    = oc1;
  ob[2 * (size_t)HWpix + p] = oc2;
}

extern "C" void kernel_launch(void* const* d_in, const int* in_sizes, int n_in,
                              void* d_out, int out_size, void* d_ws, size_t ws_size,
                              hipStream_t stream) {
  const float* raw = (const float*)d_in[0];
  const float* ys  = (const float*)d_in[1];
  const float* A   = (const float*)d_in[2];
  float* out = (float*)d_out;
  float* ws  = (float*)d_ws;

  const int B  = in_sizes[2] / (3 * NAX);        // A is [B,3,8]
  const int HW = in_sizes[0] / (3 * B);          // raw is [B,3,H,W]

  spline_setup_kernel<<<dim3(B), dim3(32), 0, stream>>>(ys, A, ws);

  const int blocks = (HW + 255) / 256;           // 256 pixels per block
  spline_main_kernel<<<dim3(blocks, B), dim3(256), 0, stream>>>(raw, ws, out, HW);
}